// MultiheadAttention_5059471475115
// MI455X (gfx1250) — compile-verified
//
#include <hip/hip_runtime.h>

typedef float v2f __attribute__((ext_vector_type(2)));
typedef float v8f __attribute__((ext_vector_type(8)));

#define NHEADS 4
#define DH 8
#define CDIM 32
#define EPS 1e-5f

// Wave-uniform broadcast of lane `l` (compile-time constant): v_readlane_b32
// into an SGPR, consumed as a scalar FMA operand. No LDS round-trip.
__device__ __forceinline__ float bcast(float v, int l) {
  return __int_as_float(__builtin_amdgcn_readlane(__float_as_int(v), l));
}

// Lane swap with lane^16 as a pure-VALU op: v_permlanex16_b32 with identity
// nibble selects exchanges lane i of row0 with lane i of row1 (wave32).
__device__ __forceinline__ float swap16(float v) {
  int i = __float_as_int(v);
  int r = __builtin_amdgcn_permlanex16(i, i, 0x76543210, 0xfedcba98, false, false);
  return __int_as_float(r);
}

__device__ __forceinline__ float wsum32(float v) {
  v += __shfl_xor(v, 16);
  v += __shfl_xor(v, 8);
  v += __shfl_xor(v, 4);
  v += __shfl_xor(v, 2);
  v += __shfl_xor(v, 1);
  return v;
}

// ---------------------------------------------------------------------------
// Stage 1: q/k/v projections. One token per 32-lane wave (lane = channel).
// Memory-bound; x values broadcast via readlane (VALU), fully coalesced I/O.
// ---------------------------------------------------------------------------
__global__ void proj_qkv_kernel(const float* __restrict__ x,
                                const float* __restrict__ wq, const float* __restrict__ bq,
                                const float* __restrict__ wk, const float* __restrict__ bk,
                                const float* __restrict__ wv, const float* __restrict__ bv,
                                float* __restrict__ qo, float* __restrict__ ko,
                                float* __restrict__ vo, int ntok) {
  const int lane = threadIdx.x & 31;
  const int tok = blockIdx.x * (blockDim.x >> 5) + (threadIdx.x >> 5);
  if (tok >= ntok) return;
  const size_t off = (size_t)tok * CDIM + lane;
  const float xv = x[off];
  float aq = bq[lane], ak = bk[lane], av = bv[lane];
#pragma unroll
  for (int j = 0; j < CDIM; ++j) {
    const float xj = bcast(xv, j);
    aq = fmaf(xj, wq[j * CDIM + lane], aq);
    ak = fmaf(xj, wk[j * CDIM + lane], ak);
    av = fmaf(xj, wv[j * CDIM + lane], av);
  }
  qo[off] = aq;
  ko[off] = ak;
  vo[off] = av;
}

// ---------------------------------------------------------------------------
// Stage 2: flash attention with fp32 WMMA (V_WMMA_F32_16X16X4_F32).
// One wave per (b, h, 16-query tile). Scores computed transposed
// (S^T = K * Q^T) so each lane owns one query column -> cheap online softmax.
// O^T accumulated via V^T * P^T, dh padded 8->16 with zero rows.
// Softmax runs in the log2 domain: scale*log2(e) folded into the Q fragment,
// so p = exp2(s - m) maps straight onto v_exp_f32 with no pre-multiply.
// All cross-lane traffic is v_permlanex16 (VALU) -- zero LDS in the hot loop.
// ---------------------------------------------------------------------------
__global__ void attn_kernel(const float* __restrict__ q, const float* __restrict__ k,
                            const float* __restrict__ v, float* __restrict__ concat,
                            int S) {
  const int lane = threadIdx.x & 31;
  const int l16 = lane & 15;
  const int half = lane >> 4;          // 0: lanes 0-15, 1: lanes 16-31
  const int gid = blockIdx.x * (blockDim.x >> 5) + (threadIdx.x >> 5);
  const int qtiles = S >> 4;
  const int bh = gid / qtiles;
  const int qt = gid - bh * qtiles;
  const int b = bh >> 2;               // / NHEADS
  const int h = bh & (NHEADS - 1);
  const int qbase = qt << 4;
  // 1/sqrt(INPUT_SIZE/NUM_HEADS) * log2(e): softmax computed base-2.
  const float scale = 0.3535533905932738f * 1.4426950408889634f;

  // B operand for S^T = K * Q^T : Q^T is (dh x 16 queries), two 4-chunks.
  // B layout (4x16, 2 VGPRs): vgpr r, lane l -> row K = r + (half?2:0), col N = l16.
  const float* qrow = q + ((size_t)b * S + qbase + l16) * CDIM + h * DH;
  v2f qB[2];
#pragma unroll
  for (int t = 0; t < 2; ++t) {
    qB[t].x = qrow[4 * t + 0 + 2 * half] * scale;
    qB[t].y = qrow[4 * t + 1 + 2 * half] * scale;
  }

  const float* kbh = k + (size_t)b * S * CDIM + h * DH;
  const float* vbh = v + (size_t)b * S * CDIM + h * DH;

  v8f acc = {};          // O^T accumulator: vgpr r -> dh = r + 8*half (half1 stays 0)
  float m = -1e30f;      // running max of log2-logits (per query = per lane)
  float lsum = 0.0f;     // running softmax denominator

  for (int j = 0; j < S; j += 16) {
    // A operand: K tile (16 keys x dh). A layout (16x4): lane -> M = l16 (key),
    // vgpr r -> K = r + (half?2:0) within the 4-chunk.
    const float* krow = kbh + (size_t)(j + l16) * CDIM;
    v2f aK[2];
#pragma unroll
    for (int t = 0; t < 2; ++t) {
      aK[t].x = krow[4 * t + 0 + 2 * half];
      aK[t].y = krow[4 * t + 1 + 2 * half];
    }

    // S^T (16 keys x 16 queries): D layout vgpr r, lane l ->
    //   key = j + r + 8*half, query = qbase + l16   (log2-scaled)
    v8f s = {};
    s = __builtin_amdgcn_wmma_f32_16x16x4_f32(false, aK[0], false, qB[0],
                                              (short)0, s, false, false);
    s = __builtin_amdgcn_wmma_f32_16x16x4_f32(false, aK[1], false, qB[1],
                                              (short)0, s, false, false);

    // Online softmax update (per query = per lane; merge the two halves).
    float mt = s[0];
#pragma unroll
    for (int r = 1; r < 8; ++r) mt = fmaxf(mt, s[r]);
    mt = fmaxf(mt, swap16(mt));
    const float mnew = fmaxf(m, mt);
    const float alpha = __builtin_amdgcn_exp2f(m - mnew);

    float p[8];
    float psum = 0.0f;
#pragma unroll
    for (int r = 0; r < 8; ++r) {
      p[r] = __builtin_amdgcn_exp2f(s[r] - mnew);
      psum += p[r];
    }
    psum += swap16(psum);
    lsum = lsum * alpha + psum;
    m = mnew;
#pragma unroll
    for (int r = 0; r < 8; ++r) acc[r] *= alpha;

    // Partner exchange for the P^T B-fragments: each lane only needs 4 partner
    // values. Pre-select what to SEND so one swap serves both directions:
    //   half0 sends p[2,3,6,7] (keys 2,3,6,7  -> needed by half1)
    //   half1 sends p[0,1,4,5] (keys 8,9,12,13 -> needed by half0)
    float recv[4];
    {
      recv[0] = swap16(half ? p[0] : p[2]);
      recv[1] = swap16(half ? p[1] : p[3]);
      recv[2] = swap16(half ? p[4] : p[6]);
      recv[3] = swap16(half ? p[5] : p[7]);
    }

    // O^T += V^T * P^T, four 4-chunks over the 16 keys.
    const float* vcol = vbh + (size_t)j * CDIM + l16;  // V[j+key][dh=l16]
    const int vvalid = (l16 < DH);
#pragma unroll
    for (int t = 0; t < 4; ++t) {
      const int k0 = 4 * t + 2 * half;  // key index for vgpr component 0
      v2f aV;  // A = V^T: lane -> M = l16 = dh (rows 8..15 zero padding)
      const float v0 = vcol[(size_t)(k0 + 0) * CDIM];
      const float v1 = vcol[(size_t)(k0 + 1) * CDIM];
      aV.x = vvalid ? v0 : 0.0f;
      aV.y = vvalid ? v1 : 0.0f;
      v2f bP;  // B = P^T fragment (constant register indices only)
      if (t == 0)      { bP.x = half ? recv[0] : p[0]; bP.y = half ? recv[1] : p[1]; }
      else if (t == 1) { bP.x = half ? recv[2] : p[4]; bP.y = half ? recv[3] : p[5]; }
      else if (t == 2) { bP.x = half ? p[2] : recv[0]; bP.y = half ? p[3] : recv[1]; }
      else             { bP.x = half ? p[6] : recv[2]; bP.y = half ? p[7] : recv[3]; }
      acc = __builtin_amdgcn_wmma_f32_16x16x4_f32(false, aV, false, bP,
                                                  (short)0, acc, false, false);
    }
  }

  // Store: O^T vgpr r, lanes 0-15 hold dh = r for query = qbase + lane.
  const float inv = 1.0f / lsum;
  if (lane < 16) {
    float* orow = concat + ((size_t)b * S + qbase + lane) * CDIM + h * DH;
#pragma unroll
    for (int r = 0; r < DH; ++r) orow[r] = acc[r] * inv;
  }
}

// ---------------------------------------------------------------------------
// Stage 3: out-projection + residual + LN1 + FF + residual + LN2.
// One token per 32-lane wave (lane = channel); broadcasts via readlane,
// LN reductions via xor butterflies (wave32 exactly covers C=32).
// ---------------------------------------------------------------------------
__global__ void epilogue_kernel(const float* __restrict__ concat, const float* __restrict__ x,
                                const float* __restrict__ wo, const float* __restrict__ bo,
                                const float* __restrict__ g1, const float* __restrict__ be1,
                                const float* __restrict__ wl, const float* __restrict__ bl,
                                const float* __restrict__ g2, const float* __restrict__ be2,
                                float* __restrict__ out, int ntok) {
  const int lane = threadIdx.x & 31;
  const int tok = blockIdx.x * (blockDim.x >> 5) + (threadIdx.x >> 5);
  if (tok >= ntok) return;
  const size_t off = (size_t)tok * CDIM + lane;

  const float cv = concat[off];
  float o = bo[lane];
#pragma unroll
  for (int j = 0; j < CDIM; ++j)
    o = fmaf(bcast(cv, j), wo[j * CDIM + lane], o);

  const float hsum = o + x[off];
  const float mu1 = wsum32(hsum) * (1.0f / 32.0f);
  const float d1 = hsum - mu1;
  const float var1 = wsum32(d1 * d1) * (1.0f / 32.0f);
  const float a1 = d1 * rsqrtf(var1 + EPS) * g1[lane] + be1[lane];

  float f = bl[lane];
#pragma unroll
  for (int j = 0; j < CDIM; ++j)
    f = fmaf(bcast(a1, j), wl[j * CDIM + lane], f);

  const float h2 = f + a1;
  const float mu2 = wsum32(h2) * (1.0f / 32.0f);
  const float d2 = h2 - mu2;
  const float var2 = wsum32(d2 * d2) * (1.0f / 32.0f);
  out[off] = d2 * rsqrtf(var2 + EPS) * g2[lane] + be2[lane];
}

// ---------------------------------------------------------------------------
extern "C" void kernel_launch(void* const* d_in, const int* in_sizes, int n_in,
                              void* d_out, int out_size, void* d_ws, size_t ws_size,
                              hipStream_t stream) {
  (void)n_in; (void)out_size; (void)ws_size;
  const float* x   = (const float*)d_in[0];
  const float* wq  = (const float*)d_in[1];
  const float* bq  = (const float*)d_in[2];
  const float* wk  = (const float*)d_in[3];
  const float* bk  = (const float*)d_in[4];
  const float* wv  = (const float*)d_in[5];
  const float* bv  = (const float*)d_in[6];
  const float* wo  = (const float*)d_in[7];
  const float* bo  = (const float*)d_in[8];
  const float* g1  = (const float*)d_in[9];
  const float* be1 = (const float*)d_in[10];
  const float* wl  = (const float*)d_in[11];
  const float* bl  = (const float*)d_in[12];
  const float* g2  = (const float*)d_in[13];
  const float* be2 = (const float*)d_in[14];
  float* out = (float*)d_out;

  const int ntok = in_sizes[0] / CDIM;  // B*S = 32768
  const int S = 2048;
  const int B = ntok / S;

  // Workspace layout: q | k | v | concat, each ntok*32 floats (4 MB) = 16 MB.
  float* qb = (float*)d_ws;
  float* kb = qb + (size_t)ntok * CDIM;
  float* vb = kb + (size_t)ntok * CDIM;
  float* cb = vb + (size_t)ntok * CDIM;

  {
    const int tpb = 256;                       // 8 tokens / block
    const int grid = (ntok + 7) / 8;
    proj_qkv_kernel<<<grid, tpb, 0, stream>>>(x, wq, bq, wk, bk, wv, bv,
                                              qb, kb, vb, ntok);
  }
  {
    const int totalWaves = B * NHEADS * (S / 16);  // 8192
    const int grid = totalWaves / 4;               // 4 waves (128 thr) per block
    attn_kernel<<<grid, 128, 0, stream>>>(qb, kb, vb, cb, S);
  }
  {
    const int tpb = 256;
    const int grid = ntok / 8;
    epilogue_kernel<<<grid, tpb, 0, stream>>>(cb, x, wo, bo, g1, be1,
                                              wl, bl, g2, be2, out, ntok);
  }
}